// AELoss_70987219468816
// MI455X (gfx1250) — compile-verified
//
#include <hip/hip_runtime.h>

#define KPT 17
#define NPERSON 30
#define NB 8
#define HWC 262144

typedef __attribute__((ext_vector_type(2))) float v2f;
typedef __attribute__((ext_vector_type(8))) float v8f;

__device__ __forceinline__ float waveReduceSum(float v) {
#pragma unroll
    for (int off = 16; off > 0; off >>= 1)
        v += __shfl_xor(v, off, 32);
    return v;
}

__global__ __launch_bounds__(256) void ae_loss_kernel(const float* __restrict__ tags,
                                                      const int* __restrict__ kpi,
                                                      float* __restrict__ out) {
    __shared__ float sMean[NB][32];
    __shared__ float sValid[NB][32];
    __shared__ float sPull[NB];
    __shared__ float sPush[NB];

    const int tid  = threadIdx.x;
    const int b    = tid >> 5;   // wave == batch
    const int lane = tid & 31;   // lane == person (30 valid, 2 pad)

    const float* pred = tags + (size_t)b * KPT * HWC;
    const int2*  kp   = (const int2*)kpi + (size_t)b * NPERSON * KPT;

    float g[KPT];
    float msk[KPT];
    float cnt = 0.f, sum = 0.f;
    if (lane < NPERSON) {
#pragma unroll
        for (int k = 0; k < KPT; ++k) {
            int2 iv = kp[lane * KPT + k];         // coalesced (idx, vis) pairs
            float m = (iv.y > 0) ? 1.f : 0.f;
            float t = pred[k * HWC + iv.x];       // scattered gather
            g[k] = t; msk[k] = m;
            cnt += m; sum += m * t;
        }
    } else {
#pragma unroll
        for (int k = 0; k < KPT; ++k) { g[k] = 0.f; msk[k] = 0.f; }
    }

    const float validf = (cnt > 0.f) ? 1.f : 0.f;
    const float safe   = fmaxf(cnt, 1.f);
    const float mean   = sum / safe;

    float pull_pp = 0.f;
#pragma unroll
    for (int k = 0; k < KPT; ++k) {
        float d = g[k] - mean;
        pull_pp += msk[k] * d * d;
    }
    pull_pp /= safe;
    const float pull_contrib = validf * pull_pp;
    const float meanP        = validf * mean;     // 0-padded mean for invalid/pad lanes

    sMean[b][lane]  = meanP;
    sValid[b][lane] = validf;
    __syncthreads();

    const float n    = waveReduceSum(validf);
    const float pull = waveReduceSum(pull_contrib);

    // Push term: 30x30 pairwise exp(-(mi-mj)^2) via rank-1 WMMA outer products.
    // diff^2 = mi^2 + mj^2 - 2*mi*mj ; mi*mj from V_WMMA_F32_16X16X4_F32 tiles.
    float pushAcc = 0.f;
    const int colOff  = lane & 15;
    const int rowHalf = (lane >= 16) ? 8 : 0;
#pragma unroll
    for (int rb = 0; rb < 32; rb += 16) {
#pragma unroll
        for (int cb = 0; cb < 32; cb += 16) {
            // A (16x4): column K=0 = mu_row, other columns zero.
            v2f A;
            A.x = (lane < 16) ? sMean[b][rb + lane] : 0.f;  // K=0 (lanes 0-15), K=2 zeroed
            A.y = 0.f;                                       // K=1 / K=3
            // B (4x16): row K=0 = mu_col, other rows zero.
            v2f Bm;
            Bm.x = (lane < 16) ? sMean[b][cb + lane] : 0.f;
            Bm.y = 0.f;
            v8f C = {};
            v8f D = __builtin_amdgcn_wmma_f32_16x16x4_f32(
                false, A, false, Bm, (short)0, C, false, false);

            const int   ncol = cb + colOff;
            const float vj   = sValid[b][ncol];
            const float mj   = sMean[b][ncol];
            const float mj2  = mj * mj;
#pragma unroll
            for (int r = 0; r < 8; ++r) {
                const int   mrow = rb + r + rowHalf;      // D VGPR r -> row M
                const float vi   = sValid[b][mrow];
                const float mi   = sMean[b][mrow];
                const float d2   = fmaxf(mi * mi + mj2 - 2.f * D[r], 0.f);
                pushAcc += vi * vj * expf(-d2);
            }
        }
    }
    const float push = waveReduceSum(pushAcc);

    if (lane == 0) {
        sPull[b] = (n > 0.f) ? pull / fmaxf(n, 1.f) : 0.f;
        sPush[b] = (n > 1.f) ? push / fmaxf(n * (n - 1.f), 1.f) * 0.5f : 0.f;
    }
    __syncthreads();

    if (tid == 0) {
        float tp = 0.f, ts = 0.f;
#pragma unroll
        for (int i = 0; i < NB; ++i) { tp += sPull[i]; ts += sPush[i]; }
        out[0] = tp;   // pull.sum()
        out[1] = ts;   // push.sum()
    }
}

extern "C" void kernel_launch(void* const* d_in, const int* in_sizes, int n_in,
                              void* d_out, int out_size, void* d_ws, size_t ws_size,
                              hipStream_t stream) {
    const float* tags = (const float*)d_in[0];
    const int*   kpi  = (const int*)d_in[1];
    float*       out  = (float*)d_out;
    ae_loss_kernel<<<1, 256, 0, stream>>>(tags, kpi, out);
}